// DAGGenome_19026705121477
// MI455X (gfx1250) — compile-verified
//
#include <hip/hip_runtime.h>
#include <stdint.h>

#define N_MAX 8192
#define THREADS 1024
#define STAGE_INTS 4096   // 16 KB staging chunk

#if defined(__has_builtin)
#if __has_builtin(__builtin_amdgcn_global_load_async_to_lds_b128) && \
    __has_builtin(__builtin_amdgcn_s_wait_asynccnt)
#define USE_ASYNC 1
#endif
#endif

typedef int v4i __attribute__((vector_size(16)));
typedef __attribute__((address_space(1))) v4i gv4i;   // global (prints as __device__)
typedef __attribute__((address_space(3))) v4i lv4i;   // LDS (prints as __shared__)

__global__ __launch_bounds__(THREADS) void dag_reach_kernel(
    const int* __restrict__ left, const int* __restrict__ right,
    float* __restrict__ out, int n)
{
    __shared__ unsigned int reach[N_MAX / 32];          // 1 KB bitmap
    __shared__ unsigned short child[2][N_MAX];          // 2 x 16 KB (0xFFFF = invalid)
    __shared__ alignas(16) int stage[STAGE_INTS];       // 16 KB async staging
    __shared__ int changed;

    const int tid = (int)threadIdx.x;

    // ---- init reachability bitmap: only node 0 reachable ----
    for (int w = tid; w < N_MAX / 32; w += THREADS) reach[w] = 0u;
    if (tid == 0) { reach[0] = 1u; changed = 1; }

    // ---- stage left/right into LDS as u16 children ----
    const int* src[2] = { left, right };
#ifdef USE_ASYNC
    for (int a = 0; a < 2; ++a) {
        for (int c = 0; c < n; c += STAGE_INTS) {
            const int cnt = min(STAGE_INTS, n - c);
            // each lane DMAs 16B global -> LDS (async, ASYNCcnt-tracked)
            if (tid * 4 + 3 < cnt) {
                __builtin_amdgcn_global_load_async_to_lds_b128(
                    (gv4i*)(src[a] + c + tid * 4),
                    (lv4i*)(&stage[tid * 4]),
                    0, 0);
            }
            __builtin_amdgcn_s_wait_asynccnt(0);
            __syncthreads();
            for (int i = tid; i < cnt; i += THREADS) {
                int v = stage[i];
                child[a][c + i] =
                    (v >= 0 && v < n) ? (unsigned short)v : (unsigned short)0xFFFFu;
            }
            __syncthreads();
        }
    }
#else
    for (int a = 0; a < 2; ++a) {
        for (int i = tid; i < n; i += THREADS) {
            int v = src[a][i];
            child[a][i] =
                (v >= 0 && v < n) ? (unsigned short)v : (unsigned short)0xFFFFu;
        }
    }
#endif
    __syncthreads();

    // ---- fixed-point reachability sweeps ----
    while (changed) {             // read is after a barrier: consistent across WG
        __syncthreads();          // everyone has consumed `changed` before reset
        if (tid == 0) changed = 0;
        __syncthreads();

        int local_changed = 0;
        for (int i = tid; i < n; i += THREADS) {
            if ((reach[i >> 5] >> (i & 31)) & 1u) {
#pragma unroll
                for (int a = 0; a < 2; ++a) {
                    unsigned int c = child[a][i];
                    if (c != 0xFFFFu) {
                        unsigned int w = c >> 5;
                        unsigned int b = 1u << (c & 31);
                        if (!(reach[w] & b)) {
                            unsigned int old = atomicOr(&reach[w], b);
                            if (!(old & b)) local_changed = 1;
                        }
                    }
                }
            }
        }
        if (local_changed) changed = 1;   // benign all-write-1 race in LDS
        __syncthreads();          // all updates visible before next condition read
    }

    // ---- emit float mask ----
    for (int i = tid; i < n; i += THREADS)
        out[i] = ((reach[i >> 5] >> (i & 31)) & 1u) ? 1.0f : 0.0f;
}

extern "C" void kernel_launch(void* const* d_in, const int* in_sizes, int n_in,
                              void* d_out, int out_size, void* d_ws, size_t ws_size,
                              hipStream_t stream) {
    // input order: thresholds, rules_left, rules_right, binary_ops, left, right
    const int* left  = (const int*)d_in[4];
    const int* right = (const int*)d_in[5];
    float* out = (float*)d_out;
    int n = in_sizes[4];
    if (n > N_MAX) n = N_MAX;
    dag_reach_kernel<<<1, THREADS, 0, stream>>>(left, right, out, n);
}